// GraphConvolution_53669911331253
// MI455X (gfx1250) — compile-verified
//
#include <hip/hip_runtime.h>

#define NN 8192
#define SS 3
#define DD 32
#define BK 64
#define TSTRIDE 68   // padded LDS row stride (floats): 4*row mod 64 -> conflict-free A reads
#define TILEF (16 * TSTRIDE)          // floats per tile buffer
#define NCHUNK (SS * 16)              // 3 supports x 16 chunks of 64 along each wave's K range

typedef __attribute__((ext_vector_type(2))) float v2f;
typedef __attribute__((ext_vector_type(4))) float v4f;
typedef __attribute__((ext_vector_type(8))) float v8f;

// ---------------- Stage 1: pre[s] = x @ W[s] + b[s], written in WMMA-B fragment order ----
// Fragment index (s, k4, ot): lane L, element j holds pre[s][k4*4 + 2*(L>>4) + j][ot*16 + (L&15)]
__global__ void gcn_stage1(const float* __restrict__ x, const float* __restrict__ W,
                           const float* __restrict__ b, float* __restrict__ preSw) {
    int tid  = blockIdx.x * blockDim.x + threadIdx.x;   // 0 .. 3*2048*2*32-1
    int lane = tid & 31;
    int frag = tid >> 5;
    int ot   = frag & 1;
    int k4   = (frag >> 1) & 2047;
    int s    = frag >> 12;
    int o    = ot * 16 + (lane & 15);
    int h    = lane >> 4;

    float r[2];
#pragma unroll
    for (int j = 0; j < 2; ++j) {
        int n = k4 * 4 + 2 * h + j;
        float acc = b[s * DD + o];
#pragma unroll
        for (int i = 0; i < DD; ++i)
            acc += x[n * DD + i] * W[(s * DD + i) * DD + o];
        r[j] = acc;
    }
    v2f v; v.x = r[0]; v.y = r[1];
    *(v2f*)(preSw + (size_t)tid * 2) = v;
}

// ---------------- Stage 2: out = relu(sum_s adj[s] @ pre[s]) via V_WMMA_F32_16X16X4_F32 --
// Block: 8 waves, rows [16b,16b+16). Wave w covers K in [1024w,1024w+1024) for every s.
// adj tiles streamed HBM->LDS with double-buffered GLOBAL_LOAD_ASYNC_TO_LDS_B128 (ASYNCcnt).
__global__ void __launch_bounds__(256) gcn_stage2(const float* __restrict__ adj,
                                                  const float* __restrict__ preSw,
                                                  float* __restrict__ out) {
    __shared__ float tiles[8 * 2 * TILEF];   // 69,632 B: double-buffered per-wave adj tiles
    __shared__ float red[8 * 32 * 16];       // 16,384 B: cross-wave reduction

    const int lane  = threadIdx.x & 31;
    const int wid   = threadIdx.x >> 5;
    const int lrow  = lane & 15;
    const int lhi   = lane >> 4;
    const int mbase = blockIdx.x * 16;

    // Wave-relative LDS byte offsets for the two tile buffers (flat addr low 32 bits).
    const unsigned ldsBase = (unsigned)(uintptr_t)(&tiles[0]);
    const unsigned bufOff0 = ldsBase + (unsigned)(wid * 2) * TILEF * 4u;
    const unsigned bufOff1 = bufOff0 + TILEF * 4u;

    // Per-lane slots for the 8 async b128 loads of a 16x64 tile.
    unsigned ldsSlot[8];   // byte offset within a buffer
    size_t   gSlot[8];     // element offset within adj[s] at column k0
#pragma unroll
    for (int i = 0; i < 8; ++i) {
        int idx = i * 32 + lane;           // 0..255
        int row = idx >> 4;                // 0..15
        int c4  = idx & 15;                // 16-byte chunk
        ldsSlot[i] = (unsigned)(row * TSTRIDE + c4 * 4) * 4u;
        gSlot[i]   = (size_t)(mbase + row) * NN + (size_t)(c4 * 4);
    }

    v8f acc0 = {};   // out cols 0..15
    v8f acc1 = {};   // out cols 16..31

    // Issue the 8 async HBM->LDS loads for chunk c into buffer with byte offset bOff.
    auto issue = [&](int c, unsigned bOff) {
        int s  = c >> 4;
        int k0 = wid * 1024 + (c & 15) * BK;
        const float* adjSK = adj + (size_t)s * NN * NN + k0;
#pragma unroll
        for (int i = 0; i < 8; ++i) {
            unsigned long long g = (unsigned long long)(uintptr_t)(adjSK + gSlot[i]);
            unsigned l = bOff + ldsSlot[i];
            asm volatile("global_load_async_to_lds_b128 %0, %1, off"
                         :: "v"(l), "v"(g) : "memory");
        }
    };

    issue(0, bufOff0);

    for (int c = 0; c < NCHUNK; ++c) {
        const float* cur = tiles + (size_t)(wid * 2 + (c & 1)) * TILEF;
        if (c + 1 < NCHUNK) {
            issue(c + 1, ((c + 1) & 1) ? bufOff1 : bufOff0);
            asm volatile("s_wait_asynccnt 8" ::: "memory");   // chunk c's 8 loads complete
        } else {
            asm volatile("s_wait_asynccnt 0" ::: "memory");
        }

        const int s  = c >> 4;
        const int k0 = wid * 1024 + (c & 15) * BK;
        const float* pa = cur + lrow * TSTRIDE + 2 * lhi;
        const float* pb = preSw + ((size_t)(s * 2048 + (k0 >> 2)) * 128) + lane * 2;
#pragma unroll
        for (int k4 = 0; k4 < BK / 4; ++k4) {
            // A fragment: lane L -> adj[mbase + (L&15)][k + 2*(L>>4) + {0,1}]
            v2f a  = *(const v2f*)(pa + k4 * 4);
            // B fragments: contiguous pre-swizzled loads, constant offsets (L2-resident)
            v2f b0 = *(const v2f*)(pb + k4 * 128);
            v2f b1 = *(const v2f*)(pb + k4 * 128 + 64);
            acc0 = __builtin_amdgcn_wmma_f32_16x16x4_f32(
                false, a, false, b0, (short)0, acc0, false, false);
            acc1 = __builtin_amdgcn_wmma_f32_16x16x4_f32(
                false, a, false, b1, (short)0, acc1, false, false);
        }
        asm volatile("" ::: "memory");
    }

    // Cross-wave reduction (fixed order -> deterministic), relu, store.
#pragma unroll
    for (int i = 0; i < 8; ++i) {
        red[(wid * 32 + lane) * 16 + i]     = acc0[i];
        red[(wid * 32 + lane) * 16 + 8 + i] = acc1[i];
    }
    __syncthreads();
    if (wid == 0) {
        float sum[16];
#pragma unroll
        for (int i = 0; i < 16; ++i) sum[i] = 0.f;
#pragma unroll
        for (int w = 0; w < 8; ++w)
#pragma unroll
            for (int i = 0; i < 16; ++i)
                sum[i] += red[(w * 32 + lane) * 16 + i];
        // C/D layout: VGPR i -> row = i + 8*(lane>=16), col = lane&15
#pragma unroll
        for (int i = 0; i < 8; ++i) {
            float v0 = sum[i];     v0 = v0 > 0.f ? v0 : 0.f;
            float v1 = sum[8 + i]; v1 = v1 > 0.f ? v1 : 0.f;
            size_t row = (size_t)(mbase + i + 8 * lhi);
            out[row * DD + lrow]      = v0;
            out[row * DD + 16 + lrow] = v1;
        }
    }
}

extern "C" void kernel_launch(void* const* d_in, const int* in_sizes, int n_in,
                              void* d_out, int out_size, void* d_ws, size_t ws_size,
                              hipStream_t stream) {
    const float* x   = (const float*)d_in[0];  // [8192, 32]
    const float* adj = (const float*)d_in[1];  // [3, 8192, 8192]
    const float* W   = (const float*)d_in[2];  // [3, 32, 32]
    const float* b   = (const float*)d_in[3];  // [3, 32]
    float* out   = (float*)d_out;              // [8192, 32]
    float* preSw = (float*)d_ws;               // 3*2048*2*64 floats = 3 MB swizzled pre

    // Stage 1: 3*2048*2 fragments * 32 lanes = 393216 threads
    gcn_stage1<<<1536, 256, 0, stream>>>(x, W, b, preSw);
    // Stage 2: 512 row-blocks of 16, 8 waves each
    gcn_stage2<<<512, 256, 0, stream>>>(adj, preSw, out);
}